// ResidualLFQ_81071802679896
// MI455X (gfx1250) — compile-verified
//
#include <hip/hip_runtime.h>

typedef float v2f __attribute__((ext_vector_type(2)));
typedef float v8f __attribute__((ext_vector_type(8)));

#define DIM 512
#define CBD 14
#define NQ  8
#define NTOK (16 * 4096)
#define NTILE (NTOK / 16)
#define OUT_ELEMS (NTOK * DIM)
#define IDX_ELEMS (NTOK * NQ)

// LDS layout (float offsets)
#define OFF_BIN   0                      // 128 ksteps * 32 lanes * 2 = 8192
#define OFF_BOUT  8192                   // 32 ntiles * 4 ksteps * 32 * 2 = 8192
#define OFF_BIAS  16384                  // 512
#define OFF_XS    16896                  // 8 waves * 16 rows * 68 (pad) = 8704
#define OFF_QO    25600                  // 8 waves * 16 rows * 20 (pad) = 2560
#define SMEM_FLOATS 28160

__global__ __launch_bounds__(256) void rlfq_fused_kernel(
    const float* __restrict__ x, const float* __restrict__ Win,
    const float* __restrict__ bin, const float* __restrict__ Wout,
    const float* __restrict__ bout, float* __restrict__ out) {
  extern __shared__ float smem[];
  float* sBin  = smem + OFF_BIN;
  float* sBout = smem + OFF_BOUT;
  float* sBias = smem + OFF_BIAS;
  const int tid = threadIdx.x;

  // ---- Build WMMA-ready B fragments for W_in (K-major swizzle, zero-pad n>=14)
  for (int i = tid; i < 128 * 32; i += 256) {
    int s = i >> 5, l = i & 31;
    int n = l & 15, kof = (l >> 4) << 1;
    int k = 4 * s + kof;
    sBin[2 * i]     = (n < CBD) ? Win[k * CBD + n] : 0.f;
    sBin[2 * i + 1] = (n < CBD) ? Win[(k + 1) * CBD + n] : 0.f;
  }
  // ---- B fragments for W_out (zero-pad K rows >= 14)
  for (int i = tid; i < 32 * 4 * 32; i += 256) {
    int l = i & 31, ks = (i >> 5) & 3, nt = i >> 7;
    int n = (nt << 4) + (l & 15), kof = (l >> 4) << 1;
    int k = 4 * ks + kof;
    sBout[2 * i]     = (k < CBD)     ? Wout[k * DIM + n]       : 0.f;
    sBout[2 * i + 1] = (k + 1 < CBD) ? Wout[(k + 1) * DIM + n] : 0.f;
  }
  for (int i = tid; i < DIM; i += 256) sBias[i] = bout[i];
  if (blockIdx.x == 0 && tid < NQ) out[OUT_ELEMS + IDX_ELEMS + tid] = 0.f; // losses
  __syncthreads();

  const int lane = tid & 31;
  const int wv   = tid >> 5;
  const int m    = lane & 15;      // M-row / N-col / d-lane depending on role
  const int hi   = lane >> 4;
  const int kof  = hi << 1;        // K offset within a 4-wide K step
  float* xs  = smem + OFF_XS + wv * (16 * 68);
  float* qos = smem + OFF_QO + wv * (16 * 20);
  const float binv = (m < CBD) ? bin[m] : 0.f;
  float* idx_out = out + OUT_ELEMS;

  const int gw = blockIdx.x * 8 + wv;
  const int nw = gridDim.x * 8;
  for (int tile = gw; tile < NTILE; tile += nw) {
    const float* xt = x + (size_t)tile * (16 * DIM);

    // ============ GEMM1: h[16 tok][16 (14 valid)] = x_tile @ W_in + b_in
    v8f c1;
#pragma unroll
    for (int r = 0; r < 8; ++r) c1[r] = binv;
    for (int kc = 0; kc < 8; ++kc) {
      // stage 16x64 x-chunk to LDS, coalesced b128 loads, pad-68 rows
#pragma unroll
      for (int i = 0; i < 8; ++i) {
        int f = (i << 5) + lane;
        int row = f >> 4, col = (f & 15) << 2;
        float4 v = *(const float4*)(xt + row * DIM + (kc << 6) + col);
        *(float4*)(xs + row * 68 + col) = v;
      }
#pragma unroll
      for (int s2 = 0; s2 < 16; ++s2) {
        v2f a = *(const v2f*)(xs + m * 68 + (s2 << 2) + kof);
        v2f b = *(const v2f*)(sBin + ((((kc << 4) + s2) << 5) + lane) * 2);
        c1 = __builtin_amdgcn_wmma_f32_16x16x4_f32(false, a, false, b,
                                                   (short)0, c1, false, false);
      }
    }

    // ============ Residual LFQ: D-layout lane = d, reg r = token (r + 8*hi)
    v8f qo;
#pragma unroll
    for (int r = 0; r < 8; ++r) {
      float res = c1[r];
      float q = 0.f, sc = 1.f;
#pragma unroll
      for (int s = 0; s < NQ; ++s) {
        bool bit = res > 0.f;
        unsigned msk = __builtin_amdgcn_ballot_w32(bit);
        float qv = bit ? sc : -sc;
        q += qv;
        res -= qv;
        sc *= 0.5f;
        if (m == 0) {  // lanes 0 and 16 publish the two tokens' indices
          int token = (tile << 4) + r + (hi << 3);
          unsigned v = hi ? ((msk >> 16) & 0x3FFFu) : (msk & 0x3FFFu);
          idx_out[token * NQ + s] = (float)v;
        }
      }
      if (m >= CBD) q = 0.f;  // K-padding for GEMM2 must be exact zeros
      qo[r] = q;
    }

    // re-swizzle quantized tile: C-layout -> A-layout via LDS (pad 20)
#pragma unroll
    for (int r = 0; r < 8; ++r) qos[(r + (hi << 3)) * 20 + m] = qo[r];
    v2f a2[4];
#pragma unroll
    for (int ks = 0; ks < 4; ++ks)
      a2[ks] = *(const v2f*)(qos + m * 20 + (ks << 2) + kof);

    // ============ GEMM2: out[16 tok][512] = qo @ W_out + b_out
    float* op = out + (size_t)(tile << 4) * DIM + m;
#pragma unroll 4
    for (int nt = 0; nt < 32; ++nt) {
      float bo = sBias[(nt << 4) + m];
      v8f c2;
#pragma unroll
      for (int r = 0; r < 8; ++r) c2[r] = bo;
#pragma unroll
      for (int ks = 0; ks < 4; ++ks) {
        v2f b = *(const v2f*)(sBout + ((((nt << 2) + ks) << 5) + lane) * 2);
        c2 = __builtin_amdgcn_wmma_f32_16x16x4_f32(false, a2[ks], false, b,
                                                   (short)0, c2, false, false);
      }
#pragma unroll
      for (int r = 0; r < 8; ++r)
        op[(size_t)(r + (hi << 3)) * DIM + (nt << 4)] = c2[r];
    }
  }
}

extern "C" void kernel_launch(void* const* d_in, const int* in_sizes, int n_in,
                              void* d_out, int out_size, void* d_ws, size_t ws_size,
                              hipStream_t stream) {
  const float* x    = (const float*)d_in[0];
  const float* Win  = (const float*)d_in[1];
  const float* bin  = (const float*)d_in[2];
  const float* Wout = (const float*)d_in[3];
  const float* bout = (const float*)d_in[4];
  float* out = (float*)d_out;

  size_t shmem = SMEM_FLOATS * sizeof(float);  // ~110 KB, within 320 KB/WGP
  (void)hipFuncSetAttribute((const void*)rlfq_fused_kernel,
                            hipFuncAttributeMaxDynamicSharedMemorySize,
                            (int)shmem);
  // 512 blocks * 8 waves = 4096 waves = exactly one 16-token tile per wave
  rlfq_fused_kernel<<<512, 256, shmem, stream>>>(x, Win, bin, Wout, bout, out);
}